// Multi_Vary_Line_Loss_39479339384984
// MI455X (gfx1250) — compile-verified
//
#include <hip/hip_runtime.h>

#define NB 512
#define NL 32
#define NP 256
#define ND 2
#define ROWLEN (NP * ND)   // 512 floats per line

typedef __attribute__((ext_vector_type(2))) float v2f;
typedef __attribute__((ext_vector_type(4))) float v4f;
typedef __attribute__((ext_vector_type(8))) float v8f;

// ---------------------------------------------------------------------------
// Kernel 1: one wave32 per batch.
//  - count n = sum(line_masks[b])
//  - clf partial: sum_l -log(prob[b,l,clf]) * mask  (fixed-order wave reduce)
//  - if n > 1: Gram = P * T^T via v_wmma_f32_16x16x4_f32 (2x2 tiles, K=512),
//    cost[i][j] = |p_i|^2 + |t_j|^2 - 2*Gram[i][j]  into LDS,
//    then replicate the reference's (buggy-greedy) assignment loop exactly.
//  - write col[] assignment to workspace.
// ---------------------------------------------------------------------------
__global__ __launch_bounds__(32)
void match_kernel(const float* __restrict__ pred,
                  const float* __restrict__ tgt,
                  const float* __restrict__ lmask,
                  const float* __restrict__ prob,
                  const int*   __restrict__ clf,
                  int*   __restrict__ cols,
                  float* __restrict__ clf_part,
                  float* __restrict__ lm_part)
{
    __shared__ float s_cost[NL * 33];   // 32x32, stride 33 (avoid bank conflicts)
    __shared__ float s_u[34];           // row potentials u[0..n]
    __shared__ float s_pn[NL];
    __shared__ float s_tn[NL];

    const int b    = blockIdx.x;
    const int lane = threadIdx.x;       // 0..31

    // ---- line count ----
    const float lmv = lmask[b * NL + lane];
    const unsigned long long bal = __ballot(lmv > 0.5f);
    const int n = (int)__popcll(bal);

    // ---- clf partial (deterministic butterfly) ----
    {
        const int   c  = clf[b * NL + lane];
        const float pr = prob[(b * NL + lane) * 2 + c];
        float cs = -logf(pr) * lmv;
        float ls = lmv;
        #pragma unroll
        for (int off = 16; off; off >>= 1) {
            cs += __shfl_xor(cs, off, 32);
            ls += __shfl_xor(ls, off, 32);
        }
        if (lane == 0) { clf_part[b] = cs; lm_part[b] = ls; }
    }

    if (n > 1) {
        const float* Pb = pred + (size_t)b * NL * ROWLEN;
        const float* Tb = tgt  + (size_t)b * NL * ROWLEN;

        // ---- squared norms of each row (lane = row) ----
        float sp = 0.f, st = 0.f;
        for (int k = 0; k < ROWLEN; k += 4) {
            v4f a = *(const v4f*)(Pb + lane * ROWLEN + k);
            v4f t = *(const v4f*)(Tb + lane * ROWLEN + k);
            sp += a.x * a.x + a.y * a.y + a.z * a.z + a.w * a.w;
            st += t.x * t.x + t.y * t.y + t.z * t.z + t.w * t.w;
        }
        s_pn[lane] = sp;
        s_tn[lane] = st;
        __syncthreads();

        // ---- Gram via WMMA f32 16x16x4, 2x2 tiles over K = 512 ----
        const int nlo   = lane & 15;        // row (A) / col (B) within tile
        const int hi    = lane >> 4;        // half-wave select
        const int khalf = hi * 2;           // K sub-offset for A/B fragments

        #pragma unroll
        for (int ti = 0; ti < 2; ++ti) {
            #pragma unroll
            for (int tj = 0; tj < 2; ++tj) {
                v8f acc = {0.f, 0.f, 0.f, 0.f, 0.f, 0.f, 0.f, 0.f};
                const float* Arow = Pb + (ti * 16 + nlo) * ROWLEN + khalf;
                const float* Brow = Tb + (tj * 16 + nlo) * ROWLEN + khalf;
                for (int k = 0; k < ROWLEN; k += 4) {
                    v2f a  = *(const v2f*)(Arow + k);   // A[m][kb..kb+1]
                    v2f bb = *(const v2f*)(Brow + k);   // B[kb..kb+1][n] = T[n][kb..]
                    acc = __builtin_amdgcn_wmma_f32_16x16x4_f32(
                        false, a, false, bb, (short)0, acc, false, false);
                }
                #pragma unroll
                for (int v = 0; v < 8; ++v) {
                    const int m  = ti * 16 + v + hi * 8;   // C/D layout: M = v (+8 hi half)
                    const int nn = tj * 16 + nlo;
                    s_cost[m * 33 + nn] = s_pn[m] + s_tn[nn] - 2.0f * acc[v];
                }
            }
        }

        // ---- init potentials ----
        s_u[lane] = 0.f;
        if (lane < 2) s_u[32 + lane] = 0.f;
        __syncthreads();

        // ---- replicate the reference's assignment loop ----
        // lane j holds column (j+1): v[j+1], p[j+1]
        float v_j = 0.f;
        int   p_j = 0;
        const float INFV = 1e18f;

        for (int i = 1; i <= n; ++i) {
            int  j0     = 0;
            bool used_j = false;
            while (true) {
                if (j0 != 0 && lane == (j0 - 1)) used_j = true;   // used[j0] = true
                // i0 = p[j0]
                const int src = (j0 == 0) ? 0 : (j0 - 1);
                const int pbr = __shfl(p_j, src, 32);
                const int i0  = (j0 == 0) ? i : pbr;
                const float ui0 = s_u[i0];
                // reduced cost on free columns (minv is effectively fresh each iter)
                const bool freelane = (lane < n) && !used_j;
                const float val = freelane
                                ? (s_cost[(i0 - 1) * 33 + lane] - ui0 - v_j)
                                : INFV;
                // argmin with numpy first-index tie-break
                float bv = val;
                int   bj = lane;
                #pragma unroll
                for (int off = 16; off; off >>= 1) {
                    const float ov = __shfl_xor(bv, off, 32);
                    const int   oj = __shfl_xor(bj, off, 32);
                    if (ov < bv || (ov == bv && oj < bj)) { bv = ov; bj = oj; }
                }
                const float delta = bv;
                const int   j1    = bj + 1;
                // u[p[used]] += delta ; v[used] -= delta   (distinct addresses)
                if (used_j) { s_u[p_j] += delta; v_j -= delta; }
                if (lane == 0) s_u[i] += delta;               // p[0] = i is always "used"
                __syncthreads();
                j0 = j1;
                const int pj0 = __shfl(p_j, j0 - 1, 32);
                if (pj0 == 0) break;
            }
            // degenerate augment (way[] == 0): only terminal column reassigned
            if (lane == (j0 - 1)) p_j = i;
        }

        // col[p[j]-1] = j-1 for assigned columns; identity for padding rows
        if (lane < n) cols[b * NL + (p_j - 1)] = lane;
        else          cols[b * NL + lane]      = lane;
    } else {
        // n == 1: matched = target as-is -> identity permutation
        cols[b * NL + lane] = lane;
    }
}

// ---------------------------------------------------------------------------
// Kernel 2: masked MSE partials, one block per batch, fixed-order reduction.
// ---------------------------------------------------------------------------
__global__ __launch_bounds__(256)
void mse_kernel(const float* __restrict__ pred,
                const float* __restrict__ tgt,
                const float* __restrict__ pmask,
                const int*   __restrict__ cols,
                float* __restrict__ mse_part,
                float* __restrict__ pm_part)
{
    __shared__ float r1[256];
    __shared__ float r2[256];

    const int b   = blockIdx.x;
    const int tid = threadIdx.x;
    const float* Pb = pred  + (size_t)b * NL * ROWLEN;
    const float* Tb = tgt   + (size_t)b * NL * ROWLEN;
    const float* Mb = pmask + (size_t)b * NL * NP;
    const int*   Cb = cols  + b * NL;

    float s = 0.f, sm = 0.f;
    for (int e = tid; e < NL * NP; e += 256) {
        const int l = e >> 8;
        const int p = e & 255;
        const int cl = Cb[l];
        const v2f pv = *(const v2f*)(Pb + e * 2);
        const v2f tv = *(const v2f*)(Tb + (cl * NP + p) * 2);
        const float m = Mb[e];
        const float dx = pv.x - tv.x;
        const float dy = pv.y - tv.y;
        s  += (dx * dx + dy * dy) * m;
        sm += m;
    }
    r1[tid] = s;
    r2[tid] = sm;
    __syncthreads();
    #pragma unroll
    for (int stp = 128; stp; stp >>= 1) {
        if (tid < stp) { r1[tid] += r1[tid + stp]; r2[tid] += r2[tid + stp]; }
        __syncthreads();
    }
    if (tid == 0) { mse_part[b] = r1[0]; pm_part[b] = r2[0]; }
}

// ---------------------------------------------------------------------------
// Kernel 3: deterministic final reduce of 512 partials -> two scalars.
// ---------------------------------------------------------------------------
__global__ __launch_bounds__(512)
void finalize_kernel(const float* __restrict__ mse_part,
                     const float* __restrict__ pm_part,
                     const float* __restrict__ clf_part,
                     const float* __restrict__ lm_part,
                     float* __restrict__ out)
{
    __shared__ float a[512], bsh[512], c[512], d[512];
    const int t = threadIdx.x;
    a[t]   = mse_part[t];
    bsh[t] = pm_part[t];
    c[t]   = clf_part[t];
    d[t]   = lm_part[t];
    __syncthreads();
    #pragma unroll
    for (int stp = 256; stp; stp >>= 1) {
        if (t < stp) {
            a[t]   += a[t + stp];
            bsh[t] += bsh[t + stp];
            c[t]   += c[t + stp];
            d[t]   += d[t + stp];
        }
        __syncthreads();
    }
    if (t == 0) {
        out[0] = a[0] / bsh[0];
        out[1] = c[0] / d[0];
    }
}

// ---------------------------------------------------------------------------
extern "C" void kernel_launch(void* const* d_in, const int* in_sizes, int n_in,
                              void* d_out, int out_size, void* d_ws, size_t ws_size,
                              hipStream_t stream)
{
    (void)in_sizes; (void)n_in; (void)out_size; (void)ws_size;

    // setup_inputs order:
    // 0 point_pos_pred [B,L,P,D] f32
    // 1 line_prob_pred [B,L,2]   f32
    // 2 target_pos     [B,L,P,D] f32
    // 3 line_masks     [B,L]     f32
    // 4 clf_line       [B,L]     i32
    // 5 point_masks    [B,L,P]   f32
    const float* pred  = (const float*)d_in[0];
    const float* prob  = (const float*)d_in[1];
    const float* tgt   = (const float*)d_in[2];
    const float* lmask = (const float*)d_in[3];
    const int*   clf   = (const int*)d_in[4];
    const float* pmask = (const float*)d_in[5];
    float* out = (float*)d_out;

    char*  ws       = (char*)d_ws;
    int*   cols     = (int*)ws;                                     // NB*NL ints
    float* mse_part = (float*)(ws + (size_t)NB * NL * sizeof(int)); // NB floats
    float* pm_part  = mse_part + NB;
    float* clf_part = pm_part + NB;
    float* lm_part  = clf_part + NB;

    match_kernel<<<NB, 32, 0, stream>>>(pred, tgt, lmask, prob, clf,
                                        cols, clf_part, lm_part);
    mse_kernel<<<NB, 256, 0, stream>>>(pred, tgt, pmask, cols, mse_part, pm_part);
    finalize_kernel<<<1, 512, 0, stream>>>(mse_part, pm_part, clf_part, lm_part, out);
}